// SwitchRouter_57681410785583
// MI455X (gfx1250) — compile-verified
//
#include <hip/hip_runtime.h>
#include <math.h>

typedef __attribute__((ext_vector_type(2))) float v2f;
typedef __attribute__((ext_vector_type(8))) float v8f;

#define M_ROWS 16384   // B*S = 4*4096
#define K_DIM  2048    // H
#define N_EXP  64      // experts
#define ROWS_PER_BLOCK 32

// ---------------------------------------------------------------------------
// Kernel 1: router logits = H[M,K] x W[N,K]^T via V_WMMA_F32_16X16X4_F32.
// 4 waves per block; block handles 16 rows x all 64 experts.
// A layout (16x4 f32): lanes 0-15 = rows, VGPR0/1 = K0/K1; lanes 16-31 = K2/K3.
// B layout (4x16 f32): lanes = N columns; VGPR0 = K0 (lo half) / K2 (hi half).
// C/D layout: VGPR v -> M = v (lanes 0-15) / v+8 (lanes 16-31), N = lane%16.
// ---------------------------------------------------------------------------
__global__ __launch_bounds__(128) void router_gemm_wmma(
    const float* __restrict__ H, const float* __restrict__ W,
    float* __restrict__ logits)
{
    const int lane = threadIdx.x & 31;
    const int wave = threadIdx.x >> 5;
    const int half = lane >> 4;   // 0: lanes 0-15, 1: lanes 16-31
    const int l    = lane & 15;

    const int rowBase = blockIdx.x * 16;
    const int colBase = wave * 16;

    const float* aPtr = H + (size_t)(rowBase + l) * K_DIM + 2 * half;
    const float* bPtr = W + (size_t)(colBase + l) * K_DIM + 2 * half;

    v8f acc = {0.f, 0.f, 0.f, 0.f, 0.f, 0.f, 0.f, 0.f};

#pragma unroll 4
    for (int kb = 0; kb < K_DIM; kb += 4) {
        v2f a = *(const v2f*)(aPtr + kb);   // K = kb+2*half, kb+2*half+1 (b64 load)
        v2f b = *(const v2f*)(bPtr + kb);
        acc = __builtin_amdgcn_wmma_f32_16x16x4_f32(
            /*neg_a=*/false, a, /*neg_b=*/false, b,
            /*c_mod=*/(short)0, acc, /*reuse_a=*/false, /*reuse_b=*/false);
    }

    float* out = logits + (size_t)(rowBase + 8 * half) * N_EXP + colBase + l;
#pragma unroll
    for (int v = 0; v < 8; ++v) {
        out[(size_t)v * N_EXP] = acc[v];
    }
}

// ---------------------------------------------------------------------------
// Kernel 2: per-row softmax stats + one-hot argmax mask.
// One wave (32 lanes) per row; lane covers experts {lane, lane+32}.
// Deterministic per-block tree reduction -> per-block partials in workspace.
// ---------------------------------------------------------------------------
__global__ __launch_bounds__(ROWS_PER_BLOCK * 32) void router_softmax_stats(
    const float* __restrict__ logits, float* __restrict__ mask,
    float* __restrict__ probPart, float* __restrict__ usagePart)
{
    __shared__ float pShare[ROWS_PER_BLOCK][N_EXP];
    __shared__ int   amaxShare[ROWS_PER_BLOCK];

    const int lane = threadIdx.x & 31;
    const int wave = threadIdx.x >> 5;
    const int row  = blockIdx.x * ROWS_PER_BLOCK + wave;

    const float v0 = logits[(size_t)row * N_EXP + lane];
    const float v1 = logits[(size_t)row * N_EXP + lane + 32];

    // row max
    float m = fmaxf(v0, v1);
#pragma unroll
    for (int off = 16; off > 0; off >>= 1)
        m = fmaxf(m, __shfl_xor(m, off, 32));

    // softmax
    const float p0 = expf(v0 - m);
    const float p1 = expf(v1 - m);
    float s = p0 + p1;
#pragma unroll
    for (int off = 16; off > 0; off >>= 1)
        s += __shfl_xor(s, off, 32);
    const float inv = 1.0f / s;

    // argmax with first-index tie-break (matches jnp.argmax)
    int c = (v0 == m) ? lane : ((v1 == m) ? (lane + 32) : N_EXP);
#pragma unroll
    for (int off = 16; off > 0; off >>= 1)
        c = min(c, __shfl_xor(c, off, 32));

    mask[(size_t)row * N_EXP + lane]      = (c == lane)      ? 1.0f : 0.0f;
    mask[(size_t)row * N_EXP + lane + 32] = (c == lane + 32) ? 1.0f : 0.0f;

    pShare[wave][lane]      = p0 * inv;
    pShare[wave][lane + 32] = p1 * inv;
    if (lane == 0) amaxShare[wave] = c;
    __syncthreads();

    if (threadIdx.x < N_EXP) {
        const int e = threadIdx.x;
        float ps = 0.0f, us = 0.0f;
#pragma unroll 4
        for (int w = 0; w < ROWS_PER_BLOCK; ++w) {
            ps += pShare[w][e];                       // contiguous row -> no bank conflicts
            us += (amaxShare[w] == e) ? 1.0f : 0.0f;
        }
        probPart[(size_t)blockIdx.x * N_EXP + e]  = ps;
        usagePart[(size_t)blockIdx.x * N_EXP + e] = us;
    }
}

// ---------------------------------------------------------------------------
// Kernel 3: fold per-block partials, compute
// loss = -sum(usage * log_softmax(router_prob)) * lambda
// ---------------------------------------------------------------------------
__global__ __launch_bounds__(64) void router_loss(
    const float* __restrict__ probPart, const float* __restrict__ usagePart,
    float* __restrict__ lossOut, int nBlocks)
{
    __shared__ float prob[N_EXP];
    __shared__ float usage[N_EXP];

    const int e = threadIdx.x;
    if (e < N_EXP) {
        float ps = 0.0f, us = 0.0f;
        for (int b = 0; b < nBlocks; ++b) {
            ps += probPart[(size_t)b * N_EXP + e];
            us += usagePart[(size_t)b * N_EXP + e];
        }
        prob[e]  = ps / (float)M_ROWS;
        usage[e] = us / (float)M_ROWS;
    }
    __syncthreads();

    if (threadIdx.x == 0) {
        float mx = -INFINITY;
        for (int i = 0; i < N_EXP; ++i) mx = fmaxf(mx, prob[i]);
        float se = 0.0f;
        for (int i = 0; i < N_EXP; ++i) se += expf(prob[i] - mx);
        const float lse = logf(se) + mx;
        float loss = 0.0f;
        for (int i = 0; i < N_EXP; ++i) loss += usage[i] * (prob[i] - lse);
        lossOut[0] = -loss * 0.01f;
    }
}

// ---------------------------------------------------------------------------
extern "C" void kernel_launch(void* const* d_in, const int* in_sizes, int n_in,
                              void* d_out, int out_size, void* d_ws, size_t ws_size,
                              hipStream_t stream)
{
    (void)in_sizes; (void)n_in; (void)out_size; (void)ws_size;

    const float* H = (const float*)d_in[0];   // [4,4096,2048] f32
    const float* W = (const float*)d_in[1];   // [64,2048]     f32

    float* logits = (float*)d_out;                              // [16384,64]
    float* mask   = logits + (size_t)M_ROWS * N_EXP;            // [16384,64]
    float* loss   = mask   + (size_t)M_ROWS * N_EXP;            // [1]

    const int nBlocks2 = M_ROWS / ROWS_PER_BLOCK;               // 512
    float* probPart  = (float*)d_ws;                            // [512,64]
    float* usagePart = probPart + (size_t)nBlocks2 * N_EXP;     // [512,64]

    router_gemm_wmma<<<M_ROWS / 16, 128, 0, stream>>>(H, W, logits);
    router_softmax_stats<<<nBlocks2, ROWS_PER_BLOCK * 32, 0, stream>>>(
        logits, mask, probPart, usagePart);
    router_loss<<<1, 64, 0, stream>>>(probPart, usagePart, loss, nBlocks2);
}